// NirvanaHinge_47047071760755
// MI455X (gfx1250) — compile-verified
//
#include <hip/hip_runtime.h>
#include <math.h>

typedef __attribute__((ext_vector_type(2))) float v2f;
typedef __attribute__((ext_vector_type(8))) float v8f;

#define FEAT 128
#define NBLOCKS 1024
#define BLOCK 256

__global__ __launch_bounds__(BLOCK)
void NirvanaHinge_main_kernel(const float* __restrict__ x,
                              const int* __restrict__ labels,
                              const float* __restrict__ centers,
                              float* __restrict__ ws,
                              int batch) {
    const int lane = threadIdx.x & 31;
    const int wid  = threadIdx.x >> 5;
    const int wpb  = BLOCK >> 5;                 // waves per block = 8
    const int gwave  = blockIdx.x * wpb + wid;
    const int nwaves = gridDim.x * wpb;
    const int half = lane >> 4;                  // 0: K=0,1   1: K=2,3
    const int m    = lane & 15;                  // row within 16-row tile

    // ---- margin = ||centers[0] - centers[1]|| / 10 (redundant per wave, L2-hot)
    float msum = 0.f;
    {
        const float* c0 = centers;
        const float* c1 = centers + FEAT;
        const int b = lane * 4;
        #pragma unroll
        for (int j = 0; j < 4; ++j) {
            float df = c0[b + j] - c1[b + j];
            msum += df * df;
        }
        #pragma unroll
        for (int off = 16; off > 0; off >>= 1)
            msum += __shfl_xor(msum, off, 32);
    }
    const float margin = sqrtf(msum) * 0.1f;

    const int ntiles = batch >> 4;
    float wsum = 0.f;

    for (int tile = gwave; tile < ntiles; tile += nwaves) {
        const int row = (tile << 4) + m;
        const float* xrow = x + (size_t)row * FEAT;
        const float* crow = centers + (size_t)labels[row] * FEAT;

        // prefetch next tile of x (speculative; dropped if OOB-translated)
        {
            int pt = tile + nwaves;
            if (pt >= ntiles) pt = ntiles - 1;
            const float* nxt = x + ((size_t)((pt << 4) + m)) * FEAT + half * 64;
            __builtin_prefetch(nxt, 0, 1);
            __builtin_prefetch(nxt + 32, 0, 1);
        }

        v8f acc = {};
        // D = X_tile - C_tile (16x128); Gram = D * D^T via 32 x WMMA f32 16x16x4.
        // A 16x4 layout (lanes0-15: M=lane,K=0,1 ; lanes16-31: M=lane-16,K=2,3)
        // equals B 4x16 layout for B = A^T, so one v2f feeds both operands.
        #pragma unroll
        for (int c = 0; c < 32; ++c) {
            const int k = (c << 2) + (half << 1);
            v2f xa = *(const v2f*)(xrow + k);
            v2f ca = *(const v2f*)(crow + k);
            v2f a  = xa - ca;
            acc = __builtin_amdgcn_wmma_f32_16x16x4_f32(
                /*neg_a=*/false, a, /*neg_b=*/false, a,
                /*c_mod=*/(short)0, acc, /*reuse_a=*/false, /*reuse_b=*/false);
        }

        // Diagonal of 16x16 f32 C: element (M,M): for M<8 at VGPR M, lane M;
        // for M>=8 at VGPR M-8, lane 16+M. m<8 lanes take acc[m]: C[m][m] in lanes 0-7, C[8+m][m] in lanes 16-23.
        float dv = acc[0];
        #pragma unroll
        for (int j = 1; j < 8; ++j)
            dv = (m == j) ? acc[j] : dv;

        float h = fmaxf(dv - margin, 0.f);
        wsum += (m < 8) ? h : 0.f;
    }

    // ---- tail rows (batch % 16), scalar, wave 0 only (no WMMA under divergence)
    if (gwave == 0) {
        for (int row = (ntiles << 4) + lane; row < batch; row += 32) {
            const float* xr = x + (size_t)row * FEAT;
            const float* cr = centers + (size_t)labels[row] * FEAT;
            float d = 0.f;
            for (int k = 0; k < FEAT; ++k) {
                float df = xr[k] - cr[k];
                d += df * df;
            }
            wsum += fmaxf(d - margin, 0.f);
        }
    }

    // ---- wave reduce, then deterministic block reduce
    #pragma unroll
    for (int off = 16; off > 0; off >>= 1)
        wsum += __shfl_xor(wsum, off, 32);

    __shared__ float ssum[BLOCK >> 5];
    if (lane == 0) ssum[wid] = wsum;
    __syncthreads();
    if (threadIdx.x == 0) {
        float b = 0.f;
        for (int i = 0; i < wpb; ++i) b += ssum[i];
        ws[blockIdx.x] = b;
    }
}

__global__ __launch_bounds__(BLOCK)
void NirvanaHinge_reduce_kernel(const float* __restrict__ ws, int n,
                                float* __restrict__ out, float inv_scale) {
    __shared__ double sdata[BLOCK];
    double s = 0.0;
    for (int i = threadIdx.x; i < n; i += BLOCK)
        s += (double)ws[i];
    sdata[threadIdx.x] = s;
    __syncthreads();
    for (int off = BLOCK >> 1; off > 0; off >>= 1) {
        if (threadIdx.x < off) sdata[threadIdx.x] += sdata[threadIdx.x + off];
        __syncthreads();
    }
    if (threadIdx.x == 0)
        out[0] = (float)(sdata[0] * (double)inv_scale);
}

extern "C" void kernel_launch(void* const* d_in, const int* in_sizes, int n_in,
                              void* d_out, int out_size, void* d_ws, size_t ws_size,
                              hipStream_t stream) {
    const float* x       = (const float*)d_in[0];
    const int*   labels  = (const int*)d_in[1];
    const float* centers = (const float*)d_in[2];
    float* out = (float*)d_out;
    float* ws  = (float*)d_ws;

    const int batch = in_sizes[1];               // labels count == batch
    const float inv_scale = 1.0f / ((float)batch * 4.0f);

    NirvanaHinge_main_kernel<<<NBLOCKS, BLOCK, 0, stream>>>(x, labels, centers, ws, batch);
    NirvanaHinge_reduce_kernel<<<1, BLOCK, 0, stream>>>(ws, NBLOCKS, out, inv_scale);
}